// ModelNew_63582695850298
// MI455X (gfx1250) — compile-verified
//
#include <hip/hip_runtime.h>
#include <stdint.h>

typedef __attribute__((ext_vector_type(2))) float v2f;
typedef __attribute__((ext_vector_type(8))) float v8f;

#define NMAT 4096
#define BT   128   // C tile per block (M and N)
#define KT   16    // k-slice per LDS stage
#define APITCH 20  // LDS pitch for A tile rows (conflict-free, 16B-aligned rows)
#define BPITCH 136 // LDS pitch for B tile rows (conflict-free, 16B-aligned rows)

// LDS byte offset = low 32 bits of the generic pointer (ISA 10.2: flat->LDS
// mapping discards the upper bits; generic = aperture | lds_offset).
__device__ __forceinline__ uint32_t lds_off32(const void* p) {
    return (uint32_t)(uintptr_t)p;
}

// Async copy 16B global -> LDS (VGLOBAL encoding, tracked by ASYNCcnt).
__device__ __forceinline__ void async_b128(uint32_t lds_off, const float* g) {
    asm volatile("global_load_async_to_lds_b128 %0, %1, off"
                 :: "v"(lds_off), "v"(g) : "memory");
}

__device__ __forceinline__ void wait_async0() {
    asm volatile("s_wait_asynccnt 0" ::: "memory");
}

// Issue one KT-slice (A: 128x16, B: 16x128) straight into LDS.
// 4 x b128 per thread, fully covering both tiles, coalesced on B.
__device__ __forceinline__ void issue_tile(const float* __restrict__ A,
                                           const float* __restrict__ B,
                                           int i0, int j0, int kb, int tid,
                                           uint32_t aBase, uint32_t bBase)
{
#pragma unroll
    for (int s = 0; s < 2; ++s) {
        const int q = tid + 256 * s;
        const int ar = q >> 2, ac = (q & 3) << 2;   // A: 4 float4 per row
        async_b128(aBase + (uint32_t)(ar * APITCH + ac) * 4u,
                   A + (size_t)(i0 + ar) * NMAT + kb + ac);
        const int br = q >> 5, bc = (q & 31) << 2;  // B: 32 float4 per row
        async_b128(bBase + (uint32_t)(br * BPITCH + bc) * 4u,
                   B + (size_t)(kb + br) * NMAT + j0 + bc);
    }
}

// In-LDS triangle fix-ups: only run on the (block-uniform) boundary slices.
__device__ __forceinline__ void fixupA(float* __restrict__ Ab,
                                       int i0, int kb, int tid)
{
#pragma unroll
    for (int s = 0; s < 8; ++s) {           // 2048 elems / 256 threads
        const int e = tid + 256 * s;
        const int r = e >> 4, c = e & 15;
        if (kb + c < i0 + r) Ab[r * APITCH + c] = 0.0f;  // triu(A): k < i
    }
}

__device__ __forceinline__ void fixupB(float* __restrict__ Bb,
                                       int j0, int kb, int tid)
{
#pragma unroll
    for (int s = 0; s < 8; ++s) {
        const int e = tid + 256 * s;
        const int r = e >> 7, c = e & 127;
        if (j0 + c < kb + r) Bb[r * BPITCH + c] = 0.0f;  // triu(B): j < k
    }
}

// One KT=16 slice: 4 k-steps x (4 m-tiles x 2 n-tiles) = 32 WMMAs per wave.
__device__ __forceinline__ void compute(const float* __restrict__ Ab,
                                        const float* __restrict__ Bb,
                                        int wmBase, int wnBase,
                                        int lmod, int lhalf,
                                        v8f acc[4][2])
{
#pragma unroll
    for (int ks = 0; ks < KT; ks += 4) {
        v2f af[4];
        v2f bf[2];
        const int kcol = ks + lhalf * 2;   // K pair this half-wave owns
#pragma unroll
        for (int mt = 0; mt < 4; ++mt) {
            const int row = wmBase + mt * 16 + lmod;        // M = lane%16
            af[mt] = *reinterpret_cast<const v2f*>(Ab + row * APITCH + kcol);
        }
#pragma unroll
        for (int nt = 0; nt < 2; ++nt) {
            const int col = wnBase + nt * 16 + lmod;        // N = lane%16
            v2f b;
            b.x = Bb[(kcol + 0) * BPITCH + col];
            b.y = Bb[(kcol + 1) * BPITCH + col];
            bf[nt] = b;
        }
#pragma unroll
        for (int mt = 0; mt < 4; ++mt)
#pragma unroll
            for (int nt = 0; nt < 2; ++nt)
                acc[mt][nt] = __builtin_amdgcn_wmma_f32_16x16x4_f32(
                    false, af[mt], false, bf[nt], (short)0, acc[mt][nt],
                    false, false);
    }
}

__global__ __launch_bounds__(256) void triu_gemm_wmma_f32(
    const float* __restrict__ A, const float* __restrict__ B,
    float* __restrict__ C)
{
    const int BJ = blockIdx.x;
    const int BI = blockIdx.y;
    const int i0 = BI * BT;
    const int j0 = BJ * BT;
    const int tid = threadIdx.x;

    if (BI > BJ) {
        // Strictly below the block diagonal: every element is zero.
        const int col = (tid & 31) << 2;
        const int rb  = tid >> 5;
        const float4 z = {0.0f, 0.0f, 0.0f, 0.0f};
#pragma unroll
        for (int i = 0; i < 16; ++i) {
            const int row = rb + i * 8;
            *reinterpret_cast<float4*>(C + (size_t)(i0 + row) * NMAT + j0 + col) = z;
        }
        return;
    }

    __shared__ __align__(16) float Alds[2][BT * APITCH];
    __shared__ __align__(16) float Blds[2][KT * BPITCH];

    const uint32_t aBase[2] = { lds_off32(&Alds[0][0]), lds_off32(&Alds[1][0]) };
    const uint32_t bBase[2] = { lds_off32(&Blds[0][0]), lds_off32(&Blds[1][0]) };

    const int wave  = tid >> 5;
    const int lane  = tid & 31;
    const int lmod  = lane & 15;
    const int lhalf = lane >> 4;
    const int wm = (wave >> 2) * 64;   // wave row offset: 0 or 64
    const int wn = (wave & 3) * 32;    // wave col offset: 0,32,64,96

    v8f acc[4][2];
#pragma unroll
    for (int mt = 0; mt < 4; ++mt)
#pragma unroll
        for (int nt = 0; nt < 2; ++nt) {
            v8f z = {0.0f, 0.0f, 0.0f, 0.0f, 0.0f, 0.0f, 0.0f, 0.0f};
            acc[mt][nt] = z;
        }

    // k ranges over [i0, (BJ+1)*BT): only k with i <= k <= j can contribute.
    const int T = (BJ - BI + 1) * (BT / KT);

    issue_tile(A, B, i0, j0, i0, tid, aBase[0], bBase[0]);
    wait_async0();
    __syncthreads();

    for (int t = 0; t < T; ++t) {
        const int p = t & 1;
        const int kb = i0 + t * KT;
        const bool hasNext = (t + 1) < T;

        // Triangle fix-ups, boundary slices only (block-uniform branches).
        const bool needA = kb < i0 + BT;   // first 8 slices of each block
        const bool needB = kb >= j0;       // last 8 slices of each block
        if (needA) fixupA(&Alds[p][0], i0, kb, tid);
        if (needB) fixupB(&Blds[p][0], j0, kb, tid);
        if (needA || needB) __syncthreads();

        if (hasNext)
            issue_tile(A, B, i0, j0, kb + KT, tid, aBase[p ^ 1], bBase[p ^ 1]);

        compute(&Alds[p][0], &Blds[p][0], wm, wn, lmod, lhalf, acc);

        if (hasNext) wait_async0();
        __syncthreads();
    }

    // Write out with final triu mask (handles diagonal blocks).
#pragma unroll
    for (int mt = 0; mt < 4; ++mt) {
#pragma unroll
        for (int nt = 0; nt < 2; ++nt) {
#pragma unroll
            for (int v = 0; v < 8; ++v) {
                const int gi = i0 + wm + mt * 16 + lhalf * 8 + v;
                const int gj = j0 + wn + nt * 16 + lmod;
                C[(size_t)gi * NMAT + gj] = (gi <= gj) ? acc[mt][nt][v] : 0.0f;
            }
        }
    }
}

extern "C" void kernel_launch(void* const* d_in, const int* in_sizes, int n_in,
                              void* d_out, int out_size, void* d_ws, size_t ws_size,
                              hipStream_t stream) {
    (void)in_sizes; (void)n_in; (void)out_size; (void)d_ws; (void)ws_size;
    const float* A = (const float*)d_in[0];
    const float* B = (const float*)d_in[1];
    float* C = (float*)d_out;
    dim3 grid(NMAT / BT, NMAT / BT);
    triu_gemm_wmma_f32<<<grid, dim3(256), 0, stream>>>(A, B, C);
}